// NPairCriterion_63187558859133
// MI455X (gfx1250) — compile-verified
//
#include <hip/hip_runtime.h>

// N-pair loss for MI455X (gfx1250, wave32).
// batch table (64MB) is L2-resident; kernel is an L2-gather-bound reduction.
// One wave32 per anchor; the 16-negative x 128-dim dot block is computed with
// chained V_WMMA_F32_16X16X4_F32 (fp32 kept for exp() accuracy).

typedef float v2f __attribute__((ext_vector_type(2)));
typedef float v8f __attribute__((ext_vector_type(8)));

#define A_COUNT   65536
#define B_ROWS    131072
#define D_DIM     128
#define M_NEG     16
#define L2_W      0.005f
#define WAVES_PB  8   // 256-thread block = 8 wave32

// ---------------------------------------------------------------------------
// Kernel 1: per-anchor n-pair term, accumulated into acc[0].
// ---------------------------------------------------------------------------
__global__ __launch_bounds__(256) void npair_kernel(
    const float* __restrict__ batch,
    const int*   __restrict__ anchors,
    const int*   __restrict__ positives,
    const int*   __restrict__ negatives,
    float*       __restrict__ acc_loss)
{
    __shared__ float a_sh[WAVES_PB][D_DIM];   // anchor vector per wave (for WMMA B broadcast)

    const int lane   = threadIdx.x & 31;
    const int wave   = threadIdx.x >> 5;
    const int anchor = blockIdx.x * WAVES_PB + wave;

    const int aRow = anchors[anchor];
    const int pRow = positives[anchor];

    // Cooperative, coalesced 512B row loads: lane l covers columns 4l..4l+3.
    const float4 av = *(const float4*)(batch + (size_t)aRow * D_DIM + lane * 4);
    const float4 pv = *(const float4*)(batch + (size_t)pRow * D_DIM + lane * 4);

    // Stage anchor vector to LDS so every lane can fetch arbitrary K-chunks.
    *(float4*)(&a_sh[wave][lane * 4]) = av;

    // dot(a, p) via VALU + 5-step wave32 butterfly.
    float dap = av.x * pv.x + av.y * pv.y + av.z * pv.z + av.w * pv.w;
    #pragma unroll
    for (int off = 16; off >= 1; off >>= 1)
        dap += __shfl_xor(dap, off, 32);

    // WMMA f32 16x16x4 A-matrix layout (16x4, MxK):
    //   lanes 0-15  : M = lane,    K-pair {0,1} of the 4-chunk
    //   lanes 16-31 : M = lane-16, K-pair {2,3}
    // => lane reads neg row (lane&15) at column 4c + 2*(lane>>4), as float2.
    const int    negRow  = negatives[anchor * M_NEG + (lane & 15)];
    const float* negBase = batch + (size_t)negRow * D_DIM + 2 * (lane >> 4);
    // B-matrix (4x16) with all 16 columns = the anchor chunk => same indexing.
    const float* aBase   = &a_sh[wave][2 * (lane >> 4)];

    __syncthreads();   // a_sh visible

    // D += A(16 negs x 4) * B(4 x 16 replicated anchor); chain K over 128.
    v8f acc = {};
    #pragma unroll
    for (int c = 0; c < D_DIM / 4; ++c) {
        v2f Aop = *(const v2f*)(negBase + 4 * c);   // global_load_b64 (L2 hit)
        v2f Bop = *(const v2f*)(aBase   + 4 * c);   // ds_load_b64
        acc = __builtin_amdgcn_wmma_f32_16x16x4_f32(
            /*neg_a=*/false, Aop, /*neg_b=*/false, Bop,
            /*c_mod=*/(short)0, acc, /*reuse_a=*/false, /*reuse_b=*/false);
    }

    // C/D layout: VGPR v = row M=v (lanes 0-15) / M=v+8 (lanes 16-31); all N
    // columns identical. So lanes<16 hold dots m=0..7, lanes>=16 hold m=8..15.
    float s = 0.0f;
    #pragma unroll
    for (int v = 0; v < 8; ++v)
        s += __expf(acc[v] - dap);          // exp(a.n_m - a.p)
    s += __shfl_xor(s, 16, 32);             // all 16 negatives

    if (lane == 0)
        atomicAdd(acc_loss, log1pf(s));     // global_atomic_add_f32
}

// ---------------------------------------------------------------------------
// Kernel 2: sum of row L2 norms over the whole batch -> acc[1].
// One wave32 per row, one global atomic per block.
// ---------------------------------------------------------------------------
__global__ __launch_bounds__(256) void l2_kernel(
    const float* __restrict__ batch,
    float*       __restrict__ acc_norm)
{
    __shared__ float blockSum;
    if (threadIdx.x == 0) blockSum = 0.0f;
    __syncthreads();

    const int lane = threadIdx.x & 31;
    const int wave = threadIdx.x >> 5;
    const int row  = blockIdx.x * WAVES_PB + wave;

    const float4 v = *(const float4*)(batch + (size_t)row * D_DIM + lane * 4);
    float ss = v.x * v.x + v.y * v.y + v.z * v.z + v.w * v.w;
    #pragma unroll
    for (int off = 16; off >= 1; off >>= 1)
        ss += __shfl_xor(ss, off, 32);

    if (lane == 0)
        atomicAdd(&blockSum, sqrtf(ss));    // ds_add_f32
    __syncthreads();
    if (threadIdx.x == 0)
        atomicAdd(acc_norm, blockSum);
}

// ---------------------------------------------------------------------------
// Init / finalize (keep launch deterministic under graph capture).
// ---------------------------------------------------------------------------
__global__ void zero_kernel(float* ws) { ws[0] = 0.0f; ws[1] = 0.0f; }

__global__ void final_kernel(const float* __restrict__ ws, float* __restrict__ out)
{
    out[0] = ws[0] * (1.0f / (float)A_COUNT)
           + L2_W * (ws[1] * (1.0f / (float)B_ROWS));
}

// ---------------------------------------------------------------------------
extern "C" void kernel_launch(void* const* d_in, const int* in_sizes, int n_in,
                              void* d_out, int out_size, void* d_ws, size_t ws_size,
                              hipStream_t stream)
{
    const float* batch     = (const float*)d_in[0];
    const int*   anchors   = (const int*)d_in[1];
    const int*   positives = (const int*)d_in[2];
    const int*   negatives = (const int*)d_in[3];
    float*       ws        = (float*)d_ws;     // ws[0]=npair sum, ws[1]=norm sum
    float*       out       = (float*)d_out;

    zero_kernel<<<1, 1, 0, stream>>>(ws);
    l2_kernel<<<B_ROWS / WAVES_PB, 256, 0, stream>>>(batch, ws + 1);
    npair_kernel<<<A_COUNT / WAVES_PB, 256, 0, stream>>>(batch, anchors, positives,
                                                         negatives, ws);
    final_kernel<<<1, 1, 0, stream>>>(ws, out);
}